// GraphNeuralNetwork_85856396247983
// MI455X (gfx1250) — compile-verified
//
#include <hip/hip_runtime.h>

typedef float v2f __attribute__((ext_vector_type(2)));
typedef float v8f __attribute__((ext_vector_type(8)));

#define N_FEAT 128
#define LDS_STRIDE 132   // 128 + 4 pad: bank = (m*132+k)%64 spreads across banks

// ---------------- degree / normalization ----------------

__global__ void gcn_deg_init(float* __restrict__ deg, int n) {
    int i = blockIdx.x * blockDim.x + threadIdx.x;
    if (i < n) deg[i] = 1.0f;   // self-loop contributes 1 to in-degree
}

__global__ void gcn_deg_count(const long long* __restrict__ dst,
                              float* __restrict__ deg, int e) {
    int i = blockIdx.x * blockDim.x + threadIdx.x;
    if (i < e) atomicAdd(&deg[(int)dst[i]], 1.0f);
}

__global__ void gcn_rsqrt(const float* __restrict__ deg,
                          float* __restrict__ dis, int n) {
    int i = blockIdx.x * blockDim.x + threadIdx.x;
    if (i < n) {
        float d = deg[i];
        dis[i] = (d > 0.0f) ? rsqrtf(d) : 0.0f;
    }
}

// ---------------- dense GEMM via V_WMMA_F32_16X16X4_F32 ----------------
// C[n x 128] = A[n x 128] @ B[128 x 128], fp32 row-major (exact wrt reference).
// Block = 256 threads = 8 waves. All waves share one 16-row A stripe (staged
// once into LDS); wave w computes the 16x16 tile at columns [16w, 16w+16).
__global__ void gcn_gemm_wmma(const float* __restrict__ A,
                              const float* __restrict__ B,
                              float* __restrict__ C, int nrows) {
    __shared__ float As[16 * LDS_STRIDE];

    const int tid  = threadIdx.x;
    const int wave = tid >> 5;             // 0..7 -> column tile
    const int lane = tid & 31;
    const int row0 = blockIdx.x * 16;
    const int col0 = wave * 16;
    const int m    = lane & 15;            // M for A-frag, N for B-frag, col for C
    const int half = lane >> 4;            // selects K pair (A/B), row half (C)

    // Cooperative stage of the 16x128 A tile: 256 threads x 8 floats (2x b128).
    {
        const int r  = tid >> 4;           // 0..15
        const int c0 = (tid & 15) * 8;     // 8 floats per thread
        int ar = row0 + r;
        if (ar >= nrows) ar = nrows - 1;   // clamp (never hit for N%16==0)
        const float4* gp = (const float4*)(A + (size_t)ar * N_FEAT + c0);
        float4 v0 = gp[0];
        float4 v1 = gp[1];
        float* lp = &As[r * LDS_STRIDE + c0];   // 16B-aligned (132*4 and c0*4 both %16==0)
        ((float4*)lp)[0] = v0;
        ((float4*)lp)[1] = v1;
    }
    __syncthreads();

    v8f acc = {};
    const float* __restrict__ arow = &As[m * LDS_STRIDE];
    #pragma unroll 4
    for (int k0 = 0; k0 < N_FEAT; k0 += 4) {
        const int ka = k0 + half * 2;      // lanes 0-15: K=k0,k0+1; lanes 16-31: K=k0+2,k0+3
        v2f a, b;
        a.x = arow[ka];                    // ds_load_b64, conflict-free via padding
        a.y = arow[ka + 1];
        b.x = B[(size_t)ka * N_FEAT + col0 + m];
        b.y = B[(size_t)(ka + 1) * N_FEAT + col0 + m];
        // 8 args: (neg_a, A, neg_b, B, c_mod, C, reuse_a, reuse_b)
        acc = __builtin_amdgcn_wmma_f32_16x16x4_f32(
            false, a, false, b, (short)0, acc, false, false);
    }

    // C/D layout: VGPR r -> row r (lanes 0-15) / r+8 (lanes 16-31), col = lane&15
    const int crow0 = row0 + half * 8;
    float* __restrict__ cp = C + (size_t)crow0 * N_FEAT + col0 + m;
    if (row0 + 16 <= nrows) {              // fast path: whole tile in-bounds
        #pragma unroll
        for (int r = 0; r < 8; ++r) cp[(size_t)r * N_FEAT] = acc[r];
    } else {
        #pragma unroll
        for (int r = 0; r < 8; ++r)
            if (crow0 + r < nrows) cp[(size_t)r * N_FEAT] = acc[r];
    }
}

// ---------------- aggregation ----------------

// out[i,f] = h[i,f] * dis[i]^2   (self-loop term; also initializes out)
__global__ void gcn_self_init(const float* __restrict__ h,
                              const float* __restrict__ dis,
                              float* __restrict__ out, long long total) {
    long long idx = (long long)blockIdx.x * blockDim.x + threadIdx.x;
    if (idx < total) {
        int i = (int)(idx >> 7);           // / 128
        float s = dis[i];
        out[idx] = h[idx] * (s * s);
    }
}

// One wave per edge: out[dst, :] += h[src, :] * dis[src]*dis[dst]
// Each lane moves a float4 (32 lanes * 4 = 128 features): coalesced gather
// from L2-resident h, then 4 global_atomic_add_f32 per lane.
__global__ void gcn_edge_scatter(const long long* __restrict__ src,
                                 const long long* __restrict__ dst,
                                 const float* __restrict__ dis,
                                 const float* __restrict__ h,
                                 float* __restrict__ out, int e) {
    const int widx = (int)(((long long)blockIdx.x * blockDim.x + threadIdx.x) >> 5);
    const int lane = threadIdx.x & 31;
    if (widx >= e) return;
    const int s = (int)src[widx];
    const int d = (int)dst[widx];
    const float norm = dis[s] * dis[d];
    const float4 v = ((const float4*)(h + (size_t)s * N_FEAT))[lane];
    float* od = out + (size_t)d * N_FEAT + lane * 4;
    atomicAdd(od + 0, v.x * norm);
    atomicAdd(od + 1, v.y * norm);
    atomicAdd(od + 2, v.z * norm);
    atomicAdd(od + 3, v.w * norm);
}

__global__ void gcn_bias_relu(float* __restrict__ out,
                              const float* __restrict__ b, long long total) {
    long long idx = (long long)blockIdx.x * blockDim.x + threadIdx.x;
    if (idx < total) {
        float v = out[idx] + b[idx & (N_FEAT - 1)];
        out[idx] = v > 0.0f ? v : 0.0f;
    }
}

// ---------------- launch ----------------

extern "C" void kernel_launch(void* const* d_in, const int* in_sizes, int n_in,
                              void* d_out, int out_size, void* d_ws, size_t ws_size,
                              hipStream_t stream) {
    (void)n_in; (void)out_size; (void)ws_size;

    const float*     x  = (const float*)d_in[0];
    const long long* ei = (const long long*)d_in[1];   // int64 [2, E] flat
    const float*     W1 = (const float*)d_in[2];
    const float*     b1 = (const float*)d_in[3];
    const float*     W2 = (const float*)d_in[4];
    const float*     b2 = (const float*)d_in[5];
    float*           out = (float*)d_out;

    const int N = in_sizes[0] / N_FEAT;        // 100000
    const int E = in_sizes[1] / 2;             // 1600000
    const long long ND = (long long)N * N_FEAT;

    const long long* src = ei;
    const long long* dst = ei + E;

    // workspace layout (floats)
    float* h    = (float*)d_ws;                // [N,128] gemm result (reused)
    float* agg1 = h + ND;                      // [N,128] layer-1 activation
    float* deg  = agg1 + ND;                   // [N]
    float* dis  = deg + N;                     // [N]

    const int BLK = 256;
    const int nBlkN    = (N + BLK - 1) / BLK;
    const int nBlkE    = (E + BLK - 1) / BLK;
    const int nBlkND   = (int)((ND + BLK - 1) / BLK);
    const int nBlkGemm = (N + 15) / 16;                               // 6250
    const int nBlkEdge = (int)(((long long)E * 32 + BLK - 1) / BLK);  // wave per edge

    // normalization (shared by both layers)
    gcn_deg_init <<<nBlkN, BLK, 0, stream>>>(deg, N);
    gcn_deg_count<<<nBlkE, BLK, 0, stream>>>(dst, deg, E);
    gcn_rsqrt    <<<nBlkN, BLK, 0, stream>>>(deg, dis, N);

    // ---- layer 1 ----
    gcn_gemm_wmma   <<<nBlkGemm, BLK, 0, stream>>>(x, W1, h, N);
    gcn_self_init   <<<nBlkND, BLK, 0, stream>>>(h, dis, agg1, ND);
    gcn_edge_scatter<<<nBlkEdge, BLK, 0, stream>>>(src, dst, dis, h, agg1, E);
    gcn_bias_relu   <<<nBlkND, BLK, 0, stream>>>(agg1, b1, ND);

    // ---- layer 2 ----
    gcn_gemm_wmma   <<<nBlkGemm, BLK, 0, stream>>>(agg1, W2, h, N);
    gcn_self_init   <<<nBlkND, BLK, 0, stream>>>(h, dis, out, ND);
    gcn_edge_scatter<<<nBlkEdge, BLK, 0, stream>>>(src, dst, dis, h, out, E);
    gcn_bias_relu   <<<nBlkND, BLK, 0, stream>>>(out, b2, ND);
}